// RealAgnosticInteractionBlock_25735444038119
// MI455X (gfx1250) — compile-verified
//
#include <hip/hip_runtime.h>
#include <cstdint>
#include <cstddef>

typedef __attribute__((ext_vector_type(2))) float v2f;
typedef __attribute__((ext_vector_type(8))) float v8f;

// D = A(16x4 f32) * B(4x16 f32) + C(16x16 f32)   -- CDNA5 V_WMMA_F32_16X16X4_F32
#define WMMA_F32X4(a, b, c) \
  __builtin_amdgcn_wmma_f32_16x16x4_f32(false, (a), false, (b), (short)0, (c), false, false)

__constant__ int d_OFFS[5] = {0, 1, 4, 9, 16};
__constant__ int d_LD[4]   = {1, 3, 5, 7};

__device__ __forceinline__ float silu(float x) { return x / (1.0f + __expf(-x)); }

__device__ __forceinline__ void atomAddF(float* p, float v) {
  __hip_atomic_fetch_add(p, v, __ATOMIC_RELAXED, __HIP_MEMORY_SCOPE_AGENT);
}

// Low 32 bits of a generic pointer to a __shared__ object == LDS byte offset
// (flat LDS aperture maps addr[31:0] directly; DS/async VGPR addresses match).
__device__ __forceinline__ uint32_t lds_off(const void* p) {
  return (uint32_t)(uintptr_t)p;
}

// CDNA5 async copy global -> LDS, 16B per lane, tracked by ASYNCcnt.
__device__ __forceinline__ void async_ld_b128(uint32_t lds, const void* gaddr) {
  asm volatile("global_load_async_to_lds_b128 %0, %1, off"
               :: "v"(lds), "v"(gaddr) : "memory");
}
__device__ __forceinline__ void wait_asynccnt0() {
  asm volatile("s_wait_asynccnt 0x0" ::: "memory");
}

// ---------------------------------------------------------------------------
// h = node_feat @ W1 * 1/sqrt(128).  M=N_nodes, N=128, K=128.
// One wave per 16x16 tile; block = 8 waves covers all 8 N-tiles of one M-tile.
// ---------------------------------------------------------------------------
__global__ void k_h_gemm(const float* __restrict__ A, const float* __restrict__ B,
                         float* __restrict__ H, int M) {
  int wv = threadIdx.x >> 5, lane = threadIdx.x & 31;
  int hf = lane >> 4, lr = lane & 15;
  int tm = blockIdx.x;      // M tile
  int tn = wv;              // N tile (0..7)
  if (tm * 16 >= M) return;

  const float* arow = A + (size_t)(tm * 16 + lr) * 128;
  int colB = tn * 16 + lr;
  v8f acc = {0.f, 0.f, 0.f, 0.f, 0.f, 0.f, 0.f, 0.f};
  for (int k0 = 0; k0 < 128; k0 += 4) {
    int ka = k0 + hf * 2;
    v2f a; a.x = arow[ka]; a.y = arow[ka + 1];
    v2f b; b.x = B[(size_t)ka * 128 + colB]; b.y = B[(size_t)(ka + 1) * 128 + colB];
    acc = WMMA_F32X4(a, b, acc);
  }
  const float sc = 0.08838834764831845f;  // 1/sqrt(128)
#pragma unroll
  for (int r = 0; r < 8; ++r)
    H[(size_t)(tm * 16 + r + hf * 8) * 128 + colB] = acc[r] * sc;
}

// ---------------------------------------------------------------------------
// Mc[i][z] = (W2[i] @ Wskip[i][:,z,:]) * (1/sqrt(C) * 1/20 * 1/sqrt(C*NZ))
// 40 blocks (i,z); 8 waves x 8 tiles each = 64 tiles of a 128x128 GEMM, K=128.
// ---------------------------------------------------------------------------
__global__ void k_combine(const float* __restrict__ W2, const float* __restrict__ Wsk,
                          float* __restrict__ Mc) {
  int i = blockIdx.x / 10, z = blockIdx.x % 10;
  int wv = threadIdx.x >> 5, lane = threadIdx.x & 31;
  int hf = lane >> 4, lr = lane & 15;
  const float sc = (float)(0.08838834764831845 * 0.05 * 0.027950849718747374);
  const float* A = W2 + (size_t)i * 128 * 128;
  float* O = Mc + (size_t)(i * 10 + z) * 16384;
  for (int t = wv * 8; t < wv * 8 + 8; ++t) {
    int tm = t >> 3, tn = t & 7;
    int colB = tn * 16 + lr;
    v8f acc = {0.f, 0.f, 0.f, 0.f, 0.f, 0.f, 0.f, 0.f};
    for (int k0 = 0; k0 < 128; k0 += 4) {
      int ka = k0 + hf * 2;
      v2f a; a.x = A[(size_t)(tm * 16 + lr) * 128 + ka];
             a.y = A[(size_t)(tm * 16 + lr) * 128 + ka + 1];
      v2f b; b.x = Wsk[(((size_t)i * 128 + ka) * 10 + z) * 128 + colB];
             b.y = Wsk[(((size_t)i * 128 + ka + 1) * 10 + z) * 128 + colB];
      acc = WMMA_F32X4(a, b, acc);
    }
#pragma unroll
    for (int r = 0; r < 8; ++r)
      O[(size_t)(tm * 16 + r + hf * 8) * 128 + colB] = acc[r] * sc;
  }
}

// ---------------------------------------------------------------------------
// Species bucketing (node_attr is one-hot over 10 species).
// ---------------------------------------------------------------------------
__device__ __forceinline__ int species_of(const float* __restrict__ attr, int n) {
  int z = 0;
#pragma unroll
  for (int j = 0; j < 10; ++j)
    if (attr[(size_t)n * 10 + j] > 0.5f) z = j;
  return z;
}

__global__ void k_count(const float* __restrict__ attr, int* counts, int N) {
  int n = blockIdx.x * blockDim.x + threadIdx.x;
  if (n >= N) return;
  atomicAdd(&counts[species_of(attr, n)], 1);
}

__global__ void k_scan(const int* __restrict__ counts, int* offsets, int* tilez) {
  if (threadIdx.x != 0 || blockIdx.x != 0) return;
  int off = 0;
  for (int z = 0; z < 10; ++z) {
    offsets[z] = off;
    off += ((counts[z] + 15) >> 4) << 4;   // 16-aligned buckets
  }
  offsets[10] = off;
  for (int t = 0; t < 512; ++t) tilez[t] = -1;
  for (int z = 0; z < 10; ++z) {
    int nt = (counts[z] + 15) >> 4;
    int t0 = offsets[z] >> 4;
    for (int t = 0; t < nt; ++t) tilez[t0 + t] = z;
  }
}

__global__ void k_place(const float* __restrict__ attr, const int* __restrict__ offsets,
                        int* cursors, int* list, int N) {
  int n = blockIdx.x * blockDim.x + threadIdx.x;
  if (n >= N) return;
  int z = species_of(attr, n);
  int pos = offsets[z] + atomicAdd(&cursors[z], 1);
  list[pos] = n;
}

// ---------------------------------------------------------------------------
// Fused edge kernel: radial MLP (8->64->64->64->512 with SiLU) via WMMA, then
// w[e,i,u] = tp_w * h[idx0[e],u], outer product with sh, atomic scatter to agg.
// One wave = 16 edges.  Block = 4 waves.  sh/emb tiles staged via CDNA5 async
// global->LDS copies (ASYNCcnt); LDS double-buffer for the D->A transpose.
// ---------------------------------------------------------------------------
__global__ void __launch_bounds__(128, 1) k_edge(
    const float* __restrict__ emb, const float* __restrict__ sh,
    const float* __restrict__ W0, const float* __restrict__ W1m,
    const float* __restrict__ W2m, const float* __restrict__ W3m,
    const float* __restrict__ H, const int* __restrict__ eidx,
    float* __restrict__ agg, int E) {
  __shared__ float T[2][4][16][68];     // padded rows: conflict-free transpose
  __shared__ float SHs[4][16][16];      // contiguous: filled by async b128
  __shared__ float EMBs[4][16][8];      // contiguous: filled by async b128
  int wv = threadIdx.x >> 5, lane = threadIdx.x & 31;
  int hf = lane >> 4, lr = lane & 15;
  int e0 = (blockIdx.x * 4 + wv) * 16;

  // --- async-stage this wave's sh (1KB) and emb (512B) tiles into LDS ------
  {
    uint32_t shl = lds_off(&SHs[wv][0][0]) + lane * 32;
    const char* shg = (const char*)(sh + (size_t)e0 * 16) + lane * 32;
    async_ld_b128(shl, shg);                 // bytes [0,16)  of lane chunk
    async_ld_b128(shl + 16, shg + 16);       // bytes [16,32)
    uint32_t eml = lds_off(&EMBs[wv][0][0]) + lane * 16;
    const char* emg = (const char*)(emb + (size_t)e0 * 8) + lane * 16;
    async_ld_b128(eml, emg);
  }
  // warm the cache for the dominant 64->512 layer weights (global_prefetch_b8)
  __builtin_prefetch(W3m + (size_t)lane * 512, 0, 0);

  int nrow[8];
#pragma unroll
  for (int r = 0; r < 8; ++r) nrow[r] = eidx[(size_t)(e0 + r + hf * 8) * 2];

  const float s0 = 0.3535533905932738f;  // 1/sqrt(8)
  const float s64 = 0.125f;              // 1/sqrt(64)

  wait_asynccnt0();                      // LDS tiles ready (wave-private)

  // layer 0: (16x8) @ (8x64), A-fragments from LDS
  {
    for (int tn = 0; tn < 4; ++tn) {
      int col = tn * 16 + lr;
      v8f acc = {0.f, 0.f, 0.f, 0.f, 0.f, 0.f, 0.f, 0.f};
      for (int k0 = 0; k0 < 8; k0 += 4) {
        int ka = k0 + hf * 2;
        v2f a; a.x = EMBs[wv][lr][ka]; a.y = EMBs[wv][lr][ka + 1];
        v2f b; b.x = W0[ka * 64 + col]; b.y = W0[(ka + 1) * 64 + col];
        acc = WMMA_F32X4(a, b, acc);
      }
#pragma unroll
      for (int r = 0; r < 8; ++r)
        T[0][wv][r + hf * 8][col] = silu(acc[r] * s0);
    }
  }
  __syncthreads();

  // layers 1,2: (16x64) @ (64x64), ping-pong LDS buffers
  for (int L = 0; L < 2; ++L) {
    const float* Wm = (L == 0) ? W1m : W2m;
    int sb = L, db = 1 - L;
    for (int tn = 0; tn < 4; ++tn) {
      int col = tn * 16 + lr;
      v8f acc = {0.f, 0.f, 0.f, 0.f, 0.f, 0.f, 0.f, 0.f};
      for (int k0 = 0; k0 < 64; k0 += 4) {
        int ka = k0 + hf * 2;
        v2f a; a.x = T[sb][wv][lr][ka]; a.y = T[sb][wv][lr][ka + 1];
        v2f b; b.x = Wm[ka * 64 + col]; b.y = Wm[(ka + 1) * 64 + col];
        acc = WMMA_F32X4(a, b, acc);
      }
#pragma unroll
      for (int r = 0; r < 8; ++r)
        T[db][wv][r + hf * 8][col] = silu(acc[r] * s64);
    }
    __syncthreads();
  }

  // final layer: (16x64) @ (64x512); A-fragments hoisted to registers
  v2f af[16];
#pragma unroll
  for (int kk = 0; kk < 16; ++kk) {
    af[kk].x = T[0][wv][lr][kk * 4 + hf * 2];
    af[kk].y = T[0][wv][lr][kk * 4 + hf * 2 + 1];
  }
  for (int tn = 0; tn < 32; ++tn) {
    int col = tn * 16 + lr;
    v8f acc = {0.f, 0.f, 0.f, 0.f, 0.f, 0.f, 0.f, 0.f};
#pragma unroll
    for (int kk = 0; kk < 16; ++kk) {
      int ka = kk * 4 + hf * 2;
      v2f b; b.x = W3m[(size_t)ka * 512 + col]; b.y = W3m[(size_t)(ka + 1) * 512 + col];
      acc = WMMA_F32X4(af[kk], b, acc);
    }
    int i = tn >> 3;          // 8 N-tiles per L-block (128 cols each)
    int u = col & 127;
    int d = d_LD[i], off = d_OFFS[i];
#pragma unroll
    for (int r = 0; r < 8; ++r) {
      float w = acc[r] * s64 * H[(size_t)nrow[r] * 128 + u];
      size_t base = (size_t)nrow[r] * 2048 + (size_t)128 * off + (size_t)u * d;
      for (int m = 0; m < d; ++m)
        atomAddF(&agg[base + m], w * SHs[wv][r + hf * 8][off + m]);
    }
  }
}

// ---------------------------------------------------------------------------
// Node output: for each 16-node tile (same species z), for each m-slot s:
// out[:, :, s] = agg_slice(16x128) @ Mc[i][z](128x128).  Wave handles 2 slots.
// ---------------------------------------------------------------------------
__global__ void __launch_bounds__(256, 1) k_nodeout(
    const float* __restrict__ agg, const float* __restrict__ Mc,
    const int* __restrict__ list, const int* __restrict__ tilez,
    float* __restrict__ out) {
  int t = blockIdx.x;
  int z = tilez[t];
  if (z < 0) return;
  int wv = threadIdx.x >> 5, lane = threadIdx.x & 31;
  int hf = lane >> 4, lr = lane & 15;
  int nlA = list[t * 16 + lr];
  float vmask = (nlA >= 0) ? 1.f : 0.f;
  int nA = (nlA < 0) ? 0 : nlA;
  int nlD[8];
#pragma unroll
  for (int r = 0; r < 8; ++r) nlD[r] = list[t * 16 + r + hf * 8];

  for (int ss = 0; ss < 2; ++ss) {
    int s = wv * 2 + ss;
    int i = (s < 1) ? 0 : (s < 4) ? 1 : (s < 9) ? 2 : 3;
    int off = d_OFFS[i], d = d_LD[i], m = s - off;
    size_t abase = (size_t)nA * 2048 + (size_t)128 * off + m;

    v2f af[32];
#pragma unroll
    for (int kk = 0; kk < 32; ++kk) {
      int u = kk * 4 + hf * 2;
      af[kk].x = vmask * agg[abase + (size_t)u * d];
      af[kk].y = vmask * agg[abase + (size_t)(u + 1) * d];
    }
    const float* B = Mc + (size_t)(i * 10 + z) * 16384;
    for (int tn = 0; tn < 8; ++tn) {
      int col = tn * 16 + lr;
      v8f acc = {0.f, 0.f, 0.f, 0.f, 0.f, 0.f, 0.f, 0.f};
#pragma unroll
      for (int kk = 0; kk < 32; ++kk) {
        int ka = kk * 4 + hf * 2;
        v2f b; b.x = B[(size_t)ka * 128 + col]; b.y = B[(size_t)(ka + 1) * 128 + col];
        acc = WMMA_F32X4(af[kk], b, acc);
      }
#pragma unroll
      for (int r = 0; r < 8; ++r) {
        if (nlD[r] >= 0)
          out[(size_t)nlD[r] * 2048 + (size_t)col * 16 + s] = acc[r];
      }
    }
  }
}

// ---------------------------------------------------------------------------
extern "C" void kernel_launch(void* const* d_in, const int* in_sizes, int n_in,
                              void* d_out, int out_size, void* d_ws, size_t ws_size,
                              hipStream_t stream) {
  const float* node_feat = (const float*)d_in[0];
  const float* node_attr = (const float*)d_in[1];
  const float* edge_sh   = (const float*)d_in[2];
  const float* emb       = (const float*)d_in[3];
  const float* W1        = (const float*)d_in[4];
  const float* w0        = (const float*)d_in[5];
  const float* w1        = (const float*)d_in[6];
  const float* w2        = (const float*)d_in[7];
  const float* w3        = (const float*)d_in[8];
  const float* W2        = (const float*)d_in[9];
  const float* Wskip     = (const float*)d_in[10];
  const int*   eidx      = (const int*)d_in[11];

  int N = in_sizes[0] / 128;   // 8000
  int E = in_sizes[2] / 16;    // 120000

  float* h   = (float*)d_ws;
  float* agg = h + (size_t)N * 128;
  float* Mc  = agg + (size_t)N * 2048;
  int* ib      = (int*)(Mc + (size_t)4 * 10 * 128 * 128);
  int* counts  = ib;            // 10 used
  int* offsets = ib + 16;       // 11 used
  int* cursors = ib + 32;       // 10 used
  int* list    = ib + 48;       // up to 8160 used
  int* tilez   = ib + 48 + 8192;// 512 used

  hipMemsetAsync(agg, 0, (size_t)N * 2048 * sizeof(float), stream);
  hipMemsetAsync(counts, 0, 48 * sizeof(int), stream);      // counts+offsets+cursors
  hipMemsetAsync(list, 0xFF, 8192 * sizeof(int), stream);   // -1 padding

  k_h_gemm<<<N / 16, 256, 0, stream>>>(node_feat, W1, h, N);
  k_combine<<<40, 256, 0, stream>>>(W2, Wskip, Mc);
  k_count<<<(N + 255) / 256, 256, 0, stream>>>(node_attr, counts, N);
  k_scan<<<1, 32, 0, stream>>>(counts, offsets, tilez);
  k_place<<<(N + 255) / 256, 256, 0, stream>>>(node_attr, offsets, cursors, list, N);
  k_edge<<<E / 64, 128, 0, stream>>>(emb, edge_sh, w0, w1, w2, w3, h, eidx, agg, E);
  k_nodeout<<<512, 256, 0, stream>>>(agg, Mc, list, tilez, (float*)d_out);
}